// InvariantMessage_13005160972666
// MI455X (gfx1250) — compile-verified
//
#include <hip/hip_runtime.h>
#include <hip/hip_bf16.h>

typedef float v2f __attribute__((ext_vector_type(2)));
typedef float v8f __attribute__((ext_vector_type(8)));

#define N_NODES 50000
#define N_EDGES 400000
#define FD      128
#define FD3     384
#define NRBF    20
#define CUTOFF  5.0f
#define PI_F    3.14159265358979323846f

__device__ __forceinline__ float silu_f(float x) {
    return x / (1.0f + expf(-x));
}

// D/C 16x16 f32 layout: lane L, vgpr r -> (M = 8*(L>>4) + r, N = (L&15))
// A 16x4 f32: lane L -> a.x = A[L&15][2*(L>>4)], a.y = A[L&15][2*(L>>4)+1]
// B 4x16 f32: lane L -> b.x = B[2*(L>>4)][L&15], b.y = B[2*(L>>4)+1][L&15]

// Kernel A: phi_nodes = silu(S @ W1 + b1) @ W2 + b2   (fp32 WMMA)
__global__ __launch_bounds__(256) void node_mlp_wmma(
    const float* __restrict__ S, const float* __restrict__ W1, const float* __restrict__ b1,
    const float* __restrict__ W2, const float* __restrict__ b2, float* __restrict__ P)
{
    __shared__ float Ss[16 * FD];   // 8 KB input tile
    __shared__ float Hs[16 * FD];   // 8 KB hidden tile

    const int tid  = threadIdx.x;
    const int wave = tid >> 5;
    const int lane = tid & 31;
    const int lm   = lane & 15;
    const int hi   = lane >> 4;
    const int koff = hi * 2;
    const int cm   = hi * 8;
    const int m0   = blockIdx.x * 16;   // 50000 = 16 * 3125 exactly

    for (int i = tid; i < 16 * FD; i += 256)
        Ss[i] = S[(size_t)(m0 + (i >> 7)) * FD + (i & 127)];
    __syncthreads();

    // Stage 1: H = silu(S@W1 + b1); 8 waves cover the 8 N-tiles of 128 cols
    {
        const int n0 = wave * 16;
        v8f acc;
        const float bb = b1[n0 + lm];
        #pragma unroll
        for (int r = 0; r < 8; ++r) acc[r] = bb;
        #pragma unroll 4
        for (int kc = 0; kc < FD / 4; ++kc) {
            const int k = kc * 4 + koff;
            v2f a = *reinterpret_cast<const v2f*>(&Ss[lm * FD + k]);
            v2f b;
            b.x = W1[(k    ) * FD + n0 + lm];
            b.y = W1[(k + 1) * FD + n0 + lm];
            acc = __builtin_amdgcn_wmma_f32_16x16x4_f32(false, a, false, b,
                                                        (short)0, acc, false, false);
        }
        #pragma unroll
        for (int r = 0; r < 8; ++r)
            Hs[(cm + r) * FD + n0 + lm] = silu_f(acc[r]);
    }
    __syncthreads();

    // Stage 2: P = H@W2 + b2; 24 N-tiles, 3 per wave
    for (int t = 0; t < 3; ++t) {
        const int n0 = (wave * 3 + t) * 16;
        v8f acc;
        const float bb = b2[n0 + lm];
        #pragma unroll
        for (int r = 0; r < 8; ++r) acc[r] = bb;
        #pragma unroll 4
        for (int kc = 0; kc < FD / 4; ++kc) {
            const int k = kc * 4 + koff;
            v2f a = *reinterpret_cast<const v2f*>(&Hs[lm * FD + k]);
            v2f b;
            b.x = W2[(k    ) * FD3 + n0 + lm];
            b.y = W2[(k + 1) * FD3 + n0 + lm];
            acc = __builtin_amdgcn_wmma_f32_16x16x4_f32(false, a, false, b,
                                                        (short)0, acc, false, false);
        }
        #pragma unroll
        for (int r = 0; r < 8; ++r)
            P[(size_t)(m0 + cm + r) * FD3 + n0 + lm] = acc[r];
    }
}

// Kernel B: out[e] = phi_nodes[nbrs[e,1]] * ((rbf(d_e)@Wd + bd) * env(d_e))
// 128 edges per block (16 per wave), w_s GEMM on WMMA, epilogue is the
// bandwidth-critical gather + coalesced store.
__global__ __launch_bounds__(256) void edge_msg_wmma(
    const float* __restrict__ dist, const long long* __restrict__ nbrs,
    const float* __restrict__ Wd, const float* __restrict__ bd,
    const float* __restrict__ P, float* __restrict__ out)
{
    __shared__ float Wd_s[NRBF * FD3];   // 30 KB
    __shared__ float bd_s[FD3];
    __shared__ float rbf_s[128 * NRBF];  // 10 KB
    __shared__ float env_s[128];
    __shared__ int   nbrj_s[128];

    const int tid  = threadIdx.x;
    const int wave = tid >> 5;
    const int lane = tid & 31;
    const int lm   = lane & 15;
    const int hi   = lane >> 4;
    const int koff = hi * 2;
    const int cm   = hi * 8;
    const int base = blockIdx.x * 128;   // 400000 = 128 * 3125 exactly

    for (int i = tid; i < NRBF * FD3; i += 256) Wd_s[i] = Wd[i];
    for (int i = tid; i < FD3; i += 256)        bd_s[i] = bd[i];
    if (tid < 128) {
        const int e  = base + tid;
        const float d = dist[e];
        env_s[tid]  = (d < CUTOFF) ? 0.5f * (cosf(d * (PI_F / CUTOFF)) + 1.0f) : 0.0f;
        nbrj_s[tid] = (int)nbrs[2 * (long long)e + 1];
    }
    for (int i = tid; i < 128 * NRBF; i += 256) {
        const int ei = i / NRBF;
        const int n  = i - ei * NRBF;
        const float d = dist[base + ei];
        const float s = sinf((float)(n + 1) * (PI_F / CUTOFF) * d);
        rbf_s[i] = (d == 0.0f) ? 0.0f : s / d;
    }
    __syncthreads();

    const int le = wave * 16;            // this wave's 16 edges within block
    for (int t = 0; t < 3; ++t) {
        const int n0  = (wave * 3 + t) * 16;
        const int col = n0 + lm;
        v8f acc;
        const float bb = bd_s[col];
        #pragma unroll
        for (int r = 0; r < 8; ++r) acc[r] = bb;
        #pragma unroll
        for (int kc = 0; kc < NRBF / 4; ++kc) {   // K = 20 = 5 * 4
            const int k = kc * 4 + koff;
            v2f a = *reinterpret_cast<const v2f*>(&rbf_s[(le + lm) * NRBF + k]);
            v2f b;
            b.x = Wd_s[(k    ) * FD3 + col];
            b.y = Wd_s[(k + 1) * FD3 + col];
            acc = __builtin_amdgcn_wmma_f32_16x16x4_f32(false, a, false, b,
                                                        (short)0, acc, false, false);
        }
        #pragma unroll
        for (int r = 0; r < 8; ++r) {
            const int ler = le + cm + r;
            const float ws = acc[r] * env_s[ler];
            const size_t row = (size_t)nbrj_s[ler] * FD3;
            const float p = P[row + col];
            out[(size_t)(base + ler) * FD3 + col] = p * ws;
        }
    }
}

extern "C" void kernel_launch(void* const* d_in, const int* in_sizes, int n_in,
                              void* d_out, int out_size, void* d_ws, size_t ws_size,
                              hipStream_t stream) {
    const float*     s_j  = (const float*)d_in[0];
    const float*     dist = (const float*)d_in[1];
    const long long* nbrs = (const long long*)d_in[2];   // int64 in reference
    const float*     W1   = (const float*)d_in[3];
    const float*     b1   = (const float*)d_in[4];
    const float*     W2   = (const float*)d_in[5];
    const float*     b2   = (const float*)d_in[6];
    const float*     Wd   = (const float*)d_in[7];
    const float*     bd   = (const float*)d_in[8];
    float*           outp = (float*)d_out;
    float*           phi  = (float*)d_ws;   // 50000*384 f32 = 76.8 MB scratch

    node_mlp_wmma<<<N_NODES / 16, 256, 0, stream>>>(s_j, W1, b1, W2, b2, phi);
    edge_msg_wmma<<<N_EDGES / 128, 256, 0, stream>>>(dist, nbrs, Wd, bd, phi, outp);
}